// LineGraphConv_77695958385169
// MI455X (gfx1250) — compile-verified
//
#include <hip/hip_runtime.h>
#include <hip/hip_bf16.h>

typedef __attribute__((ext_vector_type(2))) float v2f;
typedef __attribute__((ext_vector_type(8))) float v8f;

#define S_DIM 4096
#define N_DIM 128
#define L_PAD 50
#define NT 2          // 16x16 N-tiles per wave -> 32 cols per wave
#define NGROUPS (N_DIM / (16 * NT))   // 4

// ---------------------------------------------------------------------------
// Kernel 1: gather session item embeddings, mean over real length.
// Writes X0 into workspace AND initializes the output accumulator.
// ---------------------------------------------------------------------------
__global__ __launch_bounds__(128) void gather_mean_kernel(
    const float* __restrict__ emb,      // [N, 128]
    const int*   __restrict__ idx,      // [S, 50], 0 = padding
    const float* __restrict__ lens,     // [S]
    float* __restrict__ X0,             // [S, 128]
    float* __restrict__ Out)            // [S, 128]
{
    const int s = blockIdx.x;
    const int d = threadIdx.x;
    const int* row = idx + s * L_PAD;
    float acc = 0.0f;
#pragma unroll
    for (int j = 0; j < L_PAD; ++j) {
        const int id = row[j];
        if (id > 0) {
            acc += emb[(size_t)(id - 1) * N_DIM + d];
        }
    }
    const float v = acc / lens[s];
    const size_t o = (size_t)s * N_DIM + d;
    X0[o]  = v;
    Out[o] = v;
}

// ---------------------------------------------------------------------------
// Kernel 2: Y[4096,128] = Mat[4096,4096] @ X[4096,128] via V_WMMA_F32_16X16X4_F32.
// One wave per 16x32 output strip. 256 M-tiles * 4 N-groups = 1024 waves.
//
// Fragment layouts (wave32, from CDNA5 ISA 7.12.2):
//   A 16x4 f32:  lane l -> row m = l&15, k0 = 2*(l>>4); VGPR0 = A[m][k0], VGPR1 = A[m][k0+1]
//   B 4x16 f32:  lane l -> col n = l&15, k0 = 2*(l>>4); VGPR0 = B[k0][n], VGPR1 = B[k0+1][n]
//   C/D 16x16:   VGPR r -> row r (lanes 0-15) / row r+8 (lanes 16-31), col = l&15
// ---------------------------------------------------------------------------
__global__ __launch_bounds__(32) void gemm_f32_wmma(
    const float* __restrict__ Mat,   // [4096, 4096] row-major
    const float* __restrict__ X,     // [4096, 128]  row-major
    float* __restrict__ Y)           // [4096, 128]  row-major
{
    const int lane   = threadIdx.x;          // 0..31, full wave active
    const int mtile  = blockIdx.x / NGROUPS; // 0..255
    const int ngrp   = blockIdx.x % NGROUPS; // 0..3
    const int half   = lane >> 4;            // 0 or 1
    const int lane16 = lane & 15;
    const int k0     = half << 1;            // 0 or 2
    const int ncol0  = ngrp * (16 * NT);

    const float* arow = Mat + (size_t)((mtile << 4) + lane16) * S_DIM;
    const float* bcol = X + (size_t)k0 * N_DIM + ncol0 + lane16;

    v8f acc0 = {};
    v8f acc1 = {};

#pragma unroll 4
    for (int k = 0; k < S_DIM; k += 4) {
        const v2f a = *(const v2f*)(arow + k + k0);
        const float* bk = bcol + (size_t)k * N_DIM;
        v2f b0, b1;
        b0.x = bk[0];
        b0.y = bk[N_DIM];
        b1.x = bk[16];
        b1.y = bk[N_DIM + 16];
        acc0 = __builtin_amdgcn_wmma_f32_16x16x4_f32(
            false, a, false, b0, (short)0, acc0, false, false);
        acc1 = __builtin_amdgcn_wmma_f32_16x16x4_f32(
            false, a, false, b1, (short)0, acc1, false, false);
    }

    // Store D-matrix: VGPR r holds row (half*8 + r) of the 16x16 tile.
    float* yrow = Y + (size_t)((mtile << 4) + (half << 3)) * N_DIM + ncol0 + lane16;
#pragma unroll
    for (int r = 0; r < 8; ++r) {
        yrow[(size_t)r * N_DIM]      = acc0[r];
        yrow[(size_t)r * N_DIM + 16] = acc1[r];
    }
}

// ---------------------------------------------------------------------------
// Elementwise epilogues
// ---------------------------------------------------------------------------
__global__ __launch_bounds__(256) void add_kernel(
    float* __restrict__ out, const float* __restrict__ x, int n)
{
    const int i = blockIdx.x * blockDim.x + threadIdx.x;
    if (i < n) out[i] += x[i];
}

__global__ __launch_bounds__(256) void scale_kernel(
    float* __restrict__ out, float s, int n)
{
    const int i = blockIdx.x * blockDim.x + threadIdx.x;
    if (i < n) out[i] *= s;
}

// ---------------------------------------------------------------------------
// Launcher
// ---------------------------------------------------------------------------
extern "C" void kernel_launch(void* const* d_in, const int* in_sizes, int n_in,
                              void* d_out, int out_size, void* d_ws, size_t ws_size,
                              hipStream_t stream) {
    const float* Dm   = (const float*)d_in[0];   // [S, S]
    const float* Am   = (const float*)d_in[1];   // [S, S]
    const float* emb  = (const float*)d_in[2];   // [N, 128]
    const int*   idx  = (const int*)d_in[3];     // [S, 50]
    const float* lens = (const float*)d_in[4];   // [S]
    float* out = (float*)d_out;                  // [S, 128]

    const size_t xbytes = (size_t)S_DIM * N_DIM * sizeof(float);  // 2 MB
    float* buf0 = (float*)d_ws;                          // X_i
    float* buf1 = (float*)((char*)d_ws + xbytes);        // temp A@X

    const int nelem = S_DIM * N_DIM;  // 524288

    // X0 = gather-mean; out = X0
    gather_mean_kernel<<<S_DIM, 128, 0, stream>>>(emb, idx, lens, buf0, out);

    const int gemm_grid = (S_DIM / 16) * NGROUPS;  // 1024 waves

    for (int layer = 0; layer < 3; ++layer) {
        // T = A @ X_i
        gemm_f32_wmma<<<gemm_grid, 32, 0, stream>>>(Am, buf0, buf1);
        // X_{i+1} = D @ T  (== (D@A) @ X_i)
        gemm_f32_wmma<<<gemm_grid, 32, 0, stream>>>(Dm, buf1, buf0);
        // out += X_{i+1}
        add_kernel<<<(nelem + 255) / 256, 256, 0, stream>>>(out, buf0, nelem);
    }

    // mean over 4 propagation steps
    scale_kernel<<<(nelem + 255) / 256, 256, 0, stream>>>(out, 0.25f, nelem);
}